// Softcone_89129161326768
// MI455X (gfx1250) — compile-verified
//
#include <hip/hip_runtime.h>

#define SC_LAM 0.5f

constexpr int kBlocks  = 2048;   // reduction grid
constexpr int kThreads = 256;    // 8 waves per block (wave32)
constexpr int kSweeps  = 32;     // 32 sweeps x 2 exact bisection steps = 64 reference iterations

typedef __attribute__((ext_vector_type(2))) float v2f;
typedef __attribute__((ext_vector_type(4))) float v4f;
typedef __attribute__((ext_vector_type(8))) float v8f;

#if defined(__gfx1250__) && __has_builtin(__builtin_amdgcn_wmma_f32_16x16x4_f32)
#define SOFTCONE_WMMA 1
#else
#define SOFTCONE_WMMA 0
#endif

__device__ __forceinline__ float sc_tval(float num, float dv, float p) {
    // num / (p*d + lam)^2 via v_rcp_f32 (~1 ulp; only sign of sum-1 matters)
    float den = __builtin_fmaf(p, dv, SC_LAM);
    float r   = __builtin_amdgcn_rcpf(den);
    return num * (r * r);
}

// ---------------------------------------------------------------- init sweep:
// partial sums of (d*x)^2 and partial max|x|
__global__ __launch_bounds__(kThreads) void sc_reduce_init(
    const float* __restrict__ xg, const float* __restrict__ dg,
    float* __restrict__ part, int n)
{
    const int tid = threadIdx.x;
    const unsigned gid = blockIdx.x * kThreads + tid;
    const unsigned T   = (unsigned)kThreads * kBlocks;
    const v2f* __restrict__ x2 = (const v2f*)xg;
    const v2f* __restrict__ d2 = (const v2f*)dg;
    const unsigned n2      = (unsigned)n >> 1;
    const unsigned steps   = n2 / T;
    const unsigned stepsRT = (steps * 3u) >> 2;   // first 75% -> L2-resident (192MB)

    float sn = 0.0f, mx = 0.0f;
    unsigned s = 0;
    for (; s < stepsRT; ++s) {
        unsigned i = s * T + gid;
        v2f xv = x2[i], dv = d2[i];
        float a0 = dv[0] * xv[0], a1 = dv[1] * xv[1];
        sn += a0 * a0; sn += a1 * a1;
        mx = fmaxf(mx, fmaxf(fabsf(xv[0]), fabsf(xv[1])));
    }
    for (; s < steps; ++s) {                       // streaming 25%: non-temporal
        unsigned i = s * T + gid;
        v2f xv = __builtin_nontemporal_load(&x2[i]);
        v2f dv = __builtin_nontemporal_load(&d2[i]);
        float a0 = dv[0] * xv[0], a1 = dv[1] * xv[1];
        sn += a0 * a0; sn += a1 * a1;
        mx = fmaxf(mx, fmaxf(fabsf(xv[0]), fabsf(xv[1])));
    }
    for (unsigned e = steps * T * 2u + gid; e < (unsigned)n; e += T) {
        float xv = xg[e], dv = dg[e];
        float a = dv * xv;
        sn += a * a;
        mx = fmaxf(mx, fabsf(xv));
    }

    __shared__ float smS[kThreads], smM[kThreads];
    smS[tid] = sn; smM[tid] = mx; __syncthreads();
    for (int off = kThreads / 2; off; off >>= 1) {
        if (tid < off) { smS[tid] += smS[tid + off]; smM[tid] = fmaxf(smM[tid], smM[tid + off]); }
        __syncthreads();
    }
    if (tid == 0) { part[blockIdx.x] = smS[0]; part[kBlocks + blockIdx.x] = smM[0]; }
}

__global__ __launch_bounds__(kThreads) void sc_finalize_init(
    const float* __restrict__ part, float* __restrict__ state, int n)
{
    __shared__ float smS[kThreads], smM[kThreads];
    const int tid = threadIdx.x;
    float s = 0.0f, m = 0.0f;
    for (int i = tid; i < kBlocks; i += kThreads) { s += part[i]; m = fmaxf(m, part[kBlocks + i]); }
    smS[tid] = s; smM[tid] = m; __syncthreads();
    for (int off = kThreads / 2; off; off >>= 1) {
        if (tid < off) { smS[tid] += smS[tid + off]; smM[tid] = fmaxf(smM[tid], smM[tid + off]); }
        __syncthreads();
    }
    if (tid == 0) {
        state[0] = 0.0f;                     // lo
        state[1] = (float)n * smM[0];        // hi0 = N * max|x|
        state[2] = 4.0f * smS[0] - 1.0f;     // f0 = sum(num)/lam^2 - 1 (x4 is exact)
    }
}

// ------------------------------------------- one sweep = two bisection steps:
// evaluate f at the three nested fp midpoints {pl, p2, ph}; WMMA accumulates.
__global__ __launch_bounds__(kThreads) void sc_reduce_iter3(
    const float* __restrict__ xg, const float* __restrict__ dg,
    const float* __restrict__ state, float* __restrict__ part, int n)
{
    const float lo = state[0];
    const float hi = state[1];
    const float p2 = 0.5f * (lo + hi);   // reference step-1 midpoint
    const float pl = 0.5f * (lo + p2);   // step-2 midpoint if f(p2) <= 0
    const float ph = 0.5f * (p2 + hi);   // step-2 midpoint if f(p2) >  0

    const int tid = threadIdx.x;
    const unsigned gid = blockIdx.x * kThreads + tid;
    const unsigned T   = (unsigned)kThreads * kBlocks;
    const v2f* __restrict__ x2 = (const v2f*)xg;
    const v2f* __restrict__ d2 = (const v2f*)dg;
    const unsigned n2      = (unsigned)n >> 1;
    const unsigned steps   = n2 / T;
    const unsigned stepsRT = (steps * 3u) >> 2;

#if SOFTCONE_WMMA
    v8f aL = {0.f,0.f,0.f,0.f,0.f,0.f,0.f,0.f};
    v8f a2 = {0.f,0.f,0.f,0.f,0.f,0.f,0.f,0.f};
    v8f aH = {0.f,0.f,0.f,0.f,0.f,0.f,0.f,0.f};
    const v2f ones = {1.0f, 1.0f};
#endif
    float sL = 0.0f, s2 = 0.0f, sH = 0.0f;

#define SC_BODY(XV, DV)                                                     \
    do {                                                                    \
        float dx0 = (DV)[0] * (XV)[0]; float num0 = dx0 * dx0;              \
        float dx1 = (DV)[1] * (XV)[1]; float num1 = dx1 * dx1;              \
        v2f tl = { sc_tval(num0, (DV)[0], pl), sc_tval(num1, (DV)[1], pl) };\
        v2f t2 = { sc_tval(num0, (DV)[0], p2), sc_tval(num1, (DV)[1], p2) };\
        v2f th = { sc_tval(num0, (DV)[0], ph), sc_tval(num1, (DV)[1], ph) };\
        SC_ACC(tl, t2, th);                                                 \
    } while (0)

#if SOFTCONE_WMMA
#define SC_ACC(tl, t2, th)                                                            \
    do {                                                                              \
        aL = __builtin_amdgcn_wmma_f32_16x16x4_f32(false, tl, false, ones, (short)0, aL, false, false); \
        a2 = __builtin_amdgcn_wmma_f32_16x16x4_f32(false, t2, false, ones, (short)0, a2, false, false); \
        aH = __builtin_amdgcn_wmma_f32_16x16x4_f32(false, th, false, ones, (short)0, aH, false, false); \
    } while (0)
#else
#define SC_ACC(tl, t2, th)                                                  \
    do { sL += tl[0] + tl[1]; s2 += t2[0] + t2[1]; sH += th[0] + th[1]; } while (0)
#endif

    unsigned s = 0;
    for (; s < stepsRT; ++s) {                    // L2-resident region (RT policy)
        unsigned i = s * T + gid;
        v2f xv = x2[i], dv = d2[i];
        SC_BODY(xv, dv);
    }
    for (; s < steps; ++s) {                      // streaming region (non-temporal)
        unsigned i = s * T + gid;
        __builtin_prefetch(&x2[i + T], 0, 0);     // global_prefetch_b8, speculative
        __builtin_prefetch(&d2[i + T], 0, 0);
        v2f xv = __builtin_nontemporal_load(&x2[i]);
        v2f dv = __builtin_nontemporal_load(&d2[i]);
        SC_BODY(xv, dv);
    }

#if SOFTCONE_WMMA
    {   // each row-sum is replicated across 16 columns -> exact 1/16 rescale
        float rL = 0.f, r2 = 0.f, rH = 0.f;
        #pragma unroll
        for (int k = 0; k < 8; ++k) { rL += aL[k]; r2 += a2[k]; rH += aH[k]; }
        sL += rL * 0.0625f; s2 += r2 * 0.0625f; sH += rH * 0.0625f;
    }
#endif
    for (unsigned e = steps * T * 2u + gid; e < (unsigned)n; e += T) {  // tail (scalar)
        float xv = xg[e], dv = dg[e];
        float dx = dv * xv, num = dx * dx;
        sL += sc_tval(num, dv, pl);
        s2 += sc_tval(num, dv, p2);
        sH += sc_tval(num, dv, ph);
    }

    __shared__ float smL[kThreads], sm2[kThreads], smH[kThreads];
    smL[tid] = sL; sm2[tid] = s2; smH[tid] = sH; __syncthreads();
    for (int off = kThreads / 2; off; off >>= 1) {
        if (tid < off) {
            smL[tid] += smL[tid + off];
            sm2[tid] += sm2[tid + off];
            smH[tid] += smH[tid + off];
        }
        __syncthreads();
    }
    if (tid == 0) {
        part[blockIdx.x]              = smL[0];
        part[kBlocks + blockIdx.x]    = sm2[0];
        part[2 * kBlocks + blockIdx.x] = smH[0];
    }
#undef SC_BODY
#undef SC_ACC
}

__global__ __launch_bounds__(kThreads) void sc_finalize_iter3(
    const float* __restrict__ part, float* __restrict__ state)
{
    __shared__ float sm[kThreads];
    __shared__ float res[3];
    const int tid = threadIdx.x;
    for (int c = 0; c < 3; ++c) {
        float s = 0.0f;
        for (int i = tid; i < kBlocks; i += kThreads) s += part[c * kBlocks + i];
        sm[tid] = s; __syncthreads();
        for (int off = kThreads / 2; off; off >>= 1) {
            if (tid < off) sm[tid] += sm[tid + off];
            __syncthreads();
        }
        if (tid == 0) res[c] = sm[0];
        __syncthreads();
    }
    if (tid == 0) {
        float lo = state[0], hi = state[1];
        float p2 = 0.5f * (lo + hi);
        float fL = res[0] - 1.0f, f2 = res[1] - 1.0f, fH = res[2] - 1.0f;
        if (f2 > 0.0f) {                       // step 1: lo = p2
            lo = p2;
            float m2 = 0.5f * (lo + hi);       // == ph bit-exactly
            if (fH > 0.0f) lo = m2; else hi = m2;
        } else {                               // step 1: hi = p2
            hi = p2;
            float m2 = 0.5f * (lo + hi);       // == pl bit-exactly
            if (fL > 0.0f) lo = m2; else hi = m2;
        }
        state[0] = lo; state[1] = hi;
    }
}

// ---------------------------------------------------------------- output pass
__global__ __launch_bounds__(kThreads) void sc_finalize_out(
    const float* __restrict__ xg, const float* __restrict__ dg,
    const float* __restrict__ state, float* __restrict__ out, int n)
{
    const float lo = state[0], hi = state[1], f0 = state[2];
    const float alpha = 0.5f * (lo + hi);
    const bool take = (f0 > 0.0f) && (alpha > 0.0f);
    const float c = take ? (SC_LAM / alpha) : 0.0f;

    const int tid = threadIdx.x;
    const unsigned gid = blockIdx.x * kThreads + tid;
    const unsigned T   = (unsigned)kThreads * kBlocks;
    const v4f* __restrict__ x4 = (const v4f*)xg;
    const v4f* __restrict__ d4 = (const v4f*)dg;
    v4f* __restrict__ o4 = (v4f*)out;
    const unsigned n4 = (unsigned)n >> 2;
    for (unsigned i = gid; i < n4; i += T) {
        v4f y = {0.f, 0.f, 0.f, 0.f};
        if (take) {
            v4f xv = x4[i], dv = d4[i];
            #pragma unroll
            for (int k = 0; k < 4; ++k) y[k] = (dv[k] * xv[k]) / (dv[k] + c);
        }
        __builtin_nontemporal_store(y, &o4[i]);
    }
    for (unsigned e = n4 * 4u + gid; e < (unsigned)n; e += T) {
        float y = 0.0f;
        if (take) { float xv = xg[e], dv = dg[e]; y = (dv * xv) / (dv + c); }
        out[e] = y;
    }
}

extern "C" void kernel_launch(void* const* d_in, const int* in_sizes, int n_in,
                              void* d_out, int out_size, void* d_ws, size_t ws_size,
                              hipStream_t stream) {
    (void)n_in; (void)out_size; (void)ws_size;
    const float* x = (const float*)d_in[0];
    const float* d = (const float*)d_in[1];
    float* out = (float*)d_out;
    const int n = in_sizes[0];

    float* ws    = (float*)d_ws;
    float* state = ws;        // [0]=lo [1]=hi [2]=f0
    float* part  = ws + 16;   // 3*kBlocks partials (~24 KB incl. init's 2*kBlocks)

    sc_reduce_init<<<kBlocks, kThreads, 0, stream>>>(x, d, part, n);
    sc_finalize_init<<<1, kThreads, 0, stream>>>(part, state, n);
    for (int it = 0; it < kSweeps; ++it) {
        sc_reduce_iter3<<<kBlocks, kThreads, 0, stream>>>(x, d, state, part, n);
        sc_finalize_iter3<<<1, kThreads, 0, stream>>>(part, state);
    }
    sc_finalize_out<<<kBlocks, kThreads, 0, stream>>>(x, d, state, out, n);
}